// LifecycleLoss_55800215110016
// MI455X (gfx1250) — compile-verified
//
#include <hip/hip_runtime.h>
#include <hip/hip_bf16.h>
#include <math.h>

// ---------------------------------------------------------------------------
// LifecycleLoss for MI455X (gfx1250, wave32).
// ~166 MB unique bytes / call @ 23.3 TB/s  =>  ~7 us, HBM-bound.
//  * every byte read exactly once, fully coalesced (oracle row search via
//    width-16 wave shuffles instead of gather loads)
//  * cancel_logits (48 MB) read only under the `has` predicate
//  * deterministic 2-level reduction; final 16xG * ones(G) sum done with
//    V_WMMA_F32_16X16X4_F32 accumulation (exact f32, fixed order)
// ---------------------------------------------------------------------------

namespace {
constexpr int kB = 32;
constexpr int kT = 8192;
constexpr int kK = 16;
constexpr int kRows  = kB * kT;        // 262144
constexpr int kElems = kRows * kK;     // 4194304
constexpr int kGrid  = 2048;           // partial columns (multiple of 16)
constexpr int kBlock = 256;            // 8 waves / block
constexpr float kEps = 1e-7f;
}  // namespace

typedef __attribute__((ext_vector_type(2))) float v2f;
typedef __attribute__((ext_vector_type(8))) float v8f;

__device__ __forceinline__ void write_outputs(float* out, float fire_s,
                                              float valid_s, float cancel_s,
                                              float write_s, float n_live,
                                              float n_has) {
  const float n1 = fmaxf(n_live, 1.f);
  const float fire   = 1.0f * (fire_s / n1);               // LAM_FIRE
  const float cancel = (n_has > 0.f) ? (cancel_s / fmaxf(n_has, 1.f)) : 0.f;
  const float valid  = 0.5f * (valid_s / n1);              // LAM_VALID
  const float write  = 0.5f * (write_s / (float)kRows);    // LAM_WRITE * mean
  out[0] = fire;
  out[1] = cancel;
  out[2] = valid;
  out[3] = write;
  out[4] = fire + cancel + valid + write;
}

__global__ __launch_bounds__(kBlock) void lifecycle_main(
    const float* __restrict__ trig,          // (B,T,K)
    const float* __restrict__ validity,      // (B,T,K)
    const float* __restrict__ clog,          // (B,T,K,3)
    const float* __restrict__ wscore,        // (B,T,1)
    const unsigned char* __restrict__ live,  // (B,T,K) bool
    const int* __restrict__ cid,             // (B,T,K)
    const float* __restrict__ ofire,         // (B,T,K)
    const int* __restrict__ ocancel,         // (B,T,K)
    const float* __restrict__ ovalid,        // (B,T,K)
    const float* __restrict__ oshould,       // (B,T,1)
    const int* __restrict__ ocid,            // (B,T,K)
    float* __restrict__ partials)            // [16][kGrid]
{
  const int tid = threadIdx.x;
  const int stride = kGrid * kBlock;  // 524288, multiple of 16 -> 16-lane
                                      // groups stay aligned to K rows

  float fire_s = 0.f, valid_s = 0.f, cancel_s = 0.f, write_s = 0.f;
  float n_live = 0.f, n_has = 0.f;

  for (int i = blockIdx.x * kBlock + tid; i < kElems; i += stride) {
    const int row = i >> 4;

    // Speculative prefetch of the next grid-stride tile (global_prefetch_b8;
    // OOB prefetches are silently dropped).
    __builtin_prefetch(&trig[i + stride], 0, 0);
    __builtin_prefetch(&validity[i + stride], 0, 0);
    __builtin_prefetch(&ofire[i + stride], 0, 0);

    // ---- align_oracle_to_model: first j with ocid[row][j]==cid, cid!=0 ----
    const int c_id = cid[i];
    const int o_id = ocid[i];
    int jm = 16;
#pragma unroll
    for (int j = 15; j >= 0; --j) {           // downward -> keeps FIRST match
      const int oj = __shfl(o_id, j, 16);     // width-16 groups == one row
      if (oj == c_id) jm = j;
    }
    const bool found = (c_id != 0) && (jm < 16);
    const int j0 = found ? jm : 0;

    const float of_own = ofire[i];
    const float ov_own = ovalid[i];
    const int   oc_own = ocancel[i];
    float ft = __shfl(of_own, j0, 16);
    float vt = __shfl(ov_own, j0, 16);
    int   ct = __shfl(oc_own, j0, 16);
    ft = found ? ft : 0.f;
    vt = found ? vt : 0.f;
    ct = found ? ct : 0;

    const bool lv = (live[i] != 0);
    const float m = lv ? 1.f : 0.f;

    // ---- fire / valid masked BCE (clamped p => logs are finite) ----
    {
      const float p = fminf(fmaxf(trig[i], kEps), 1.f - kEps);
      const float q = fminf(fmaxf(validity[i], kEps), 1.f - kEps);
      const float bf = -(ft * logf(p) + (1.f - ft) * logf(1.f - p));
      const float bv = -(vt * logf(q) + (1.f - vt) * logf(1.f - q));
      fire_s  += m * bf;
      valid_s += m * bv;
      n_live  += m;
    }

    // ---- cancel CE over 3 logits, only where (ct>0 && live) ----
    {
      const bool has = lv && (ct > 0);
      if (has) {
        const float l0 = clog[3 * i + 0];
        const float l1 = clog[3 * i + 1];
        const float l2 = clog[3 * i + 2];
        int tgt = ct - 1;
        tgt = tgt < 0 ? 0 : (tgt > 2 ? 2 : tgt);
        const float mx  = fmaxf(l0, fmaxf(l1, l2));
        const float lse = mx + logf(expf(l0 - mx) + expf(l1 - mx) + expf(l2 - mx));
        const float lt  = (tgt == 0) ? l0 : ((tgt == 1) ? l1 : l2);
        cancel_s += (lse - lt);
        n_has += 1.f;
      }
    }

    // ---- write-head BCE-with-logits, one lane per row ----
    if ((i & 15) == 0) {
      const float x = wscore[row];
      const float y = oshould[row];
      write_s += fmaxf(x, 0.f) - x * y + log1pf(expf(-fabsf(x)));
    }
  }

  // ---- deterministic block reduction: wave shuffles, then LDS across waves
  float vals[6] = {fire_s, valid_s, cancel_s, write_s, n_live, n_has};
#pragma unroll
  for (int c = 0; c < 6; ++c) {
    float v = vals[c];
#pragma unroll
    for (int off = 16; off >= 1; off >>= 1) v += __shfl_xor(v, off, 32);
    vals[c] = v;
  }
  __shared__ float red[kBlock / 32][6];
  const int wid = tid >> 5, lane = tid & 31;
  if (lane == 0) {
#pragma unroll
    for (int c = 0; c < 6; ++c) red[wid][c] = vals[c];
  }
  __syncthreads();
  if (tid == 0) {
#pragma unroll
    for (int c = 0; c < 6; ++c) {
      float s = 0.f;
      for (int w = 0; w < kBlock / 32; ++w) s += red[w][c];
      partials[c * kGrid + blockIdx.x] = s;
    }
    // pad rows 6..15 with zeros so the WMMA finalize can consume a 16-row A
    for (int c = 6; c < 16; ++c) partials[c * kGrid + blockIdx.x] = 0.f;
  }
}

// One wave sums partials[16][kGrid] along columns: D += A(16x4) x ones(4x16).
// A f32 16x4 layout: lanes 0-15 row M, VGPR0=K0 / VGPR1=K1; lanes 16-31
// VGPR0=K2 / VGPR1=K3.  D row sums replicate across all N, so lane 0's
// c[0..5] are the six totals.
__global__ __launch_bounds__(32) void lifecycle_final(
    const float* __restrict__ partials, float* __restrict__ out) {
  const int lane = threadIdx.x;
#if defined(__has_builtin) && __has_builtin(__builtin_amdgcn_wmma_f32_16x16x4_f32)
  const int r = lane & 15;
  const int cbase = (lane >> 4) << 1;  // lanes 16-31 feed K=2,3
  v8f acc[4];
#pragma unroll
  for (int u = 0; u < 4; ++u)
    acc[u] = v8f{0.f, 0.f, 0.f, 0.f, 0.f, 0.f, 0.f, 0.f};
  const v2f ones = {1.f, 1.f};
  for (int kk = 0; kk < kGrid; kk += 16) {
#pragma unroll
    for (int u = 0; u < 4; ++u) {  // 4 independent chains hide XDL latency
      const int col = kk + 4 * u + cbase;
      const v2f a = {partials[r * kGrid + col], partials[r * kGrid + col + 1]};
      acc[u] = __builtin_amdgcn_wmma_f32_16x16x4_f32(
          false, a, false, ones, (short)0, acc[u], false, false);
    }
  }
  const v8f c = acc[0] + acc[1] + acc[2] + acc[3];
  if (lane == 0)
    write_outputs(out, c[0], c[1], c[2], c[3], c[4], c[5]);
#else
  // Deterministic scalar fallback (also what the host pass parses).
  if (lane == 0) {
    float s[6];
    for (int c = 0; c < 6; ++c) {
      float acc = 0.f;
      for (int g = 0; g < kGrid; ++g) acc += partials[c * kGrid + g];
      s[c] = acc;
    }
    write_outputs(out, s[0], s[1], s[2], s[3], s[4], s[5]);
  }
#endif
}

extern "C" void kernel_launch(void* const* d_in, const int* in_sizes, int n_in,
                              void* d_out, int out_size, void* d_ws,
                              size_t ws_size, hipStream_t stream) {
  (void)in_sizes; (void)n_in; (void)out_size; (void)ws_size;
  const float* trig           = (const float*)d_in[0];
  const float* validity       = (const float*)d_in[1];
  const float* clog           = (const float*)d_in[2];
  const float* wscore         = (const float*)d_in[3];
  const unsigned char* live   = (const unsigned char*)d_in[4];
  const int* cid              = (const int*)d_in[5];
  const float* ofire          = (const float*)d_in[6];
  const int* ocancel          = (const int*)d_in[7];
  const float* ovalid         = (const float*)d_in[8];
  const float* oshould        = (const float*)d_in[9];
  const int* ocid             = (const int*)d_in[10];
  float* partials             = (float*)d_ws;  // needs 16*kGrid*4 = 128 KB
  float* out                  = (float*)d_out;

  lifecycle_main<<<kGrid, kBlock, 0, stream>>>(
      trig, validity, clog, wscore, live, cid, ofire, ocancel, ovalid, oshould,
      ocid, partials);
  lifecycle_final<<<1, 32, 0, stream>>>(partials, out);
}